// LinearAttention_90640989815098
// MI455X (gfx1250) — compile-verified
//
#include <hip/hip_runtime.h>

typedef __attribute__((ext_vector_type(2))) float v2f;
typedef __attribute__((ext_vector_type(4))) float v4f;
typedef __attribute__((ext_vector_type(8))) float v8f;

#define BDIM 256
constexpr int B_ = 4, H_ = 16, T_ = 4096, D_ = 64;
constexpr int BH = B_ * H_;
constexpr float EPS = 1e-6f;

// pass1: TC timesteps staged per chunk, NCHUNK chunks per block
constexpr int TC = 64;
constexpr int NCHUNK = 4;
constexpr int SPLIT = T_ / (TC * NCHUNK);   // 16 blocks per (b,h)
constexpr int TCP = 68;                     // padded LDS stride (conflict-free, 16B-aligned rows)

// pass2
constexpr int TM = 64;                      // query rows per block
constexpr int SQS = 68;                     // padded LDS stride

__device__ __forceinline__ float fmap(float x) {
    // elu(x) + 1
    return x > 0.0f ? x + 1.0f : __expf(x);
}

__device__ __forceinline__ v8f wmma4(v2f a, v2f b, v8f c) {
    return __builtin_amdgcn_wmma_f32_16x16x4_f32(false, a, false, b, (short)0, c,
                                                 false, false);
}

__global__ __launch_bounds__(BDIM) void zero_ws_kernel(float* w, int n) {
    int i = blockIdx.x * BDIM + threadIdx.x;
    if (i < n) w[i] = 0.0f;
}

// Pass 1: kvT[bh][e][d] += sum_t v[t][e] * k'[t][d]   (A = v', B = k')
//         ksum[bh][d]   += sum_t k'[t][d]
__global__ __launch_bounds__(BDIM) void pass1_kv_kernel(const float* __restrict__ kin,
                                                        const float* __restrict__ vin,
                                                        const float* __restrict__ mask,
                                                        float* __restrict__ wkv,
                                                        float* __restrict__ wks) {
    __shared__ float lkT[D_ * TCP];   // k' transposed: [d][t]
    __shared__ float lvT[D_ * TCP];   // v  transposed: [e][t]

    const int tid = threadIdx.x;
    const int bh  = blockIdx.x / SPLIT;
    const int blk = blockIdx.x % SPLIT;
    const int b   = bh / H_;
    const int t00 = blk * TC * NCHUNK;

    const int wid  = tid >> 5;
    const int lane = tid & 31;
    const int half = lane >> 4;      // K-pair select within WMMA
    const int lm   = lane & 15;      // M / N index within tile
    const int quad  = wid & 3;       // which 32x32 block of the 64x64 state
    const int khalf = wid >> 2;      // which 32-timestep half of the chunk
    const int eT0 = (quad >> 1) * 32;
    const int dT0 = (quad & 1) * 32;

    v8f a00 = {}, a01 = {}, a10 = {}, a11 = {};   // acc[eSub][dSub]
    float ksr = 0.0f;
    const int dks = tid >> 2;        // ksum: 4 lanes per d-column
    const int l4  = tid & 3;

    for (int c = 0; c < NCHUNK; ++c) {
        const int t0 = t00 + c * TC;
        const size_t gbase = ((size_t)bh * T_ + t0) * D_;

        // stage transposed (b128 global loads, scalar transposed LDS stores)
        #pragma unroll
        for (int i = 0; i < (TC * D_) / (BDIM * 4); ++i) {
            int idx4 = tid + i * BDIM;
            int tl   = idx4 >> 4;          // timestep within chunk
            int d0   = (idx4 & 15) * 4;    // first of 4 feature cols
            float m  = mask[b * T_ + t0 + tl];
            v4f k4 = *(const v4f*)(kin + gbase + (size_t)idx4 * 4);
            v4f v4 = *(const v4f*)(vin + gbase + (size_t)idx4 * 4);
            #pragma unroll
            for (int j = 0; j < 4; ++j) {
                lkT[(d0 + j) * TCP + tl] = fmap(k4[j]) * m;
                lvT[(d0 + j) * TCP + tl] = v4[j] * m;
            }
        }
        __syncthreads();

        // 2x2 tile block per wave; one ds_load_b64 per fragment, 4 WMMA per iter
        #pragma unroll
        for (int kk = 0; kk < 32; kk += 4) {
            const int tr = khalf * 32 + kk + half * 2;
            v2f fa0, fa1, fb0, fb1;
            fa0.x = lvT[(eT0 + lm) * TCP + tr];
            fa0.y = lvT[(eT0 + lm) * TCP + tr + 1];
            fa1.x = lvT[(eT0 + 16 + lm) * TCP + tr];
            fa1.y = lvT[(eT0 + 16 + lm) * TCP + tr + 1];
            fb0.x = lkT[(dT0 + lm) * TCP + tr];
            fb0.y = lkT[(dT0 + lm) * TCP + tr + 1];
            fb1.x = lkT[(dT0 + 16 + lm) * TCP + tr];
            fb1.y = lkT[(dT0 + 16 + lm) * TCP + tr + 1];
            a00 = wmma4(fa0, fb0, a00);
            a01 = wmma4(fa0, fb1, a01);
            a10 = wmma4(fa1, fb0, a10);
            a11 = wmma4(fa1, fb1, a11);
        }

        // ksum partial over this chunk (contiguous LDS reads)
        {
            float s = 0.0f;
            #pragma unroll
            for (int j = 0; j < 16; ++j)
                s += lkT[dks * TCP + l4 * 16 + j];
            s += __shfl_xor(s, 1, 32);
            s += __shfl_xor(s, 2, 32);
            ksr += s;
        }
        __syncthreads();
    }

    // single atomic writeback; kvT layout [e][d] -> contiguous lanes in d
    float* kvb = wkv + (size_t)bh * D_ * D_;
    #pragma unroll
    for (int r = 0; r < 8; ++r) {
        const int e0 = eT0 + r + half * 8;
        atomicAdd(&kvb[e0 * D_ + dT0 + lm],            a00[r]);
        atomicAdd(&kvb[e0 * D_ + dT0 + 16 + lm],       a01[r]);
        atomicAdd(&kvb[(e0 + 16) * D_ + dT0 + lm],     a10[r]);
        atomicAdd(&kvb[(e0 + 16) * D_ + dT0 + 16 + lm], a11[r]);
    }
    if (l4 == 0) atomicAdd(&wks[bh * D_ + dks], ksr);
}

// Pass 2: out[t][e] = (sum_d q'[t][d] * kvT[e][d]) / (q'[t].ksum + eps)
__global__ __launch_bounds__(BDIM) void pass2_out_kernel(const float* __restrict__ qin,
                                                         const float* __restrict__ wkv,
                                                         const float* __restrict__ wks,
                                                         float* __restrict__ out) {
    __shared__ float skvT[D_ * SQS];  // [e][d]
    __shared__ float sq[TM * SQS];    // [t][d]
    __shared__ float sks[D_];
    __shared__ float sz[TM];

    const int tid   = threadIdx.x;
    const int bh    = blockIdx.x >> 6;           // T/TM = 64 chunks per bh
    const int tbase = (blockIdx.x & 63) * TM;
    const size_t qbase = ((size_t)bh * T_ + tbase) * D_;

    // stage kvT rows (coalesced b128 -> aligned b128 LDS stores)
    const float* kvg = wkv + (size_t)bh * D_ * D_;
    #pragma unroll
    for (int i = 0; i < (D_ * D_) / (BDIM * 4); ++i) {
        int idx4 = tid + i * BDIM;
        int e = idx4 >> 4, d0 = (idx4 & 15) * 4;
        *(v4f*)&skvT[e * SQS + d0] = *(const v4f*)(kvg + (size_t)idx4 * 4);
    }
    // stage q' rows
    #pragma unroll
    for (int i = 0; i < (TM * D_) / (BDIM * 4); ++i) {
        int idx4 = tid + i * BDIM;
        int row = idx4 >> 4, c0 = (idx4 & 15) * 4;
        v4f qq = *(const v4f*)(qin + qbase + (size_t)idx4 * 4);
        v4f r;
        r.x = fmap(qq.x); r.y = fmap(qq.y); r.z = fmap(qq.z); r.w = fmap(qq.w);
        *(v4f*)&sq[row * SQS + c0] = r;
    }
    if (tid < D_) sks[tid] = wks[bh * D_ + tid];
    __syncthreads();

    // z per row: 4 lanes per row, contiguous 16-wide partials, shuffle-reduce
    {
        const int row = tid >> 2;
        const int l4  = tid & 3;
        float s = 0.0f;
        #pragma unroll
        for (int j = 0; j < 16; ++j) {
            const int c = l4 * 16 + j;
            s += sq[row * SQS + c] * sks[c];
        }
        s += __shfl_xor(s, 1, 32);
        s += __shfl_xor(s, 2, 32);
        if (l4 == 0) sz[row] = s + EPS;
    }
    __syncthreads();

    const int wid  = tid >> 5;
    const int lane = tid & 31;
    const int half = lane >> 4;
    const int lm   = lane & 15;
    const int rt  = (wid >> 1) * 16;   // row-tile base (0..48)
    const int ct0 = (wid & 1) * 32;    // first col-tile base (0 or 32)

    v8f c0 = {}, c1 = {};
    #pragma unroll
    for (int kk = 0; kk < D_; kk += 4) {
        const int kr = kk + half * 2;
        v2f fa, fb0, fb1;
        fa.x  = sq[(rt + lm) * SQS + kr];
        fa.y  = sq[(rt + lm) * SQS + kr + 1];
        fb0.x = skvT[(ct0 + lm) * SQS + kr];
        fb0.y = skvT[(ct0 + lm) * SQS + kr + 1];
        fb1.x = skvT[(ct0 + 16 + lm) * SQS + kr];
        fb1.y = skvT[(ct0 + 16 + lm) * SQS + kr + 1];
        c0 = wmma4(fa, fb0, c0);
        c1 = wmma4(fa, fb1, c1);
    }
    #pragma unroll
    for (int r = 0; r < 8; ++r) {
        const int row = rt + r + half * 8;
        const float z = sz[row];
        const size_t o = ((size_t)bh * T_ + tbase + row) * D_;
        out[o + ct0 + lm]      = c0[r] / z;
        out[o + ct0 + 16 + lm] = c1[r] / z;
    }
}

extern "C" void kernel_launch(void* const* d_in, const int* in_sizes, int n_in,
                              void* d_out, int out_size, void* d_ws, size_t ws_size,
                              hipStream_t stream) {
    (void)in_sizes; (void)n_in; (void)out_size; (void)ws_size;
    const float* q    = (const float*)d_in[0];
    const float* k    = (const float*)d_in[1];
    const float* v    = (const float*)d_in[2];
    const float* mask = (const float*)d_in[3];
    float* out = (float*)d_out;
    float* wkv = (float*)d_ws;                    // BH * 64 * 64 (kvT layout [e][d])
    float* wks = wkv + (size_t)BH * D_ * D_;      // BH * 64

    const int nz = BH * D_ * D_ + BH * D_;
    zero_ws_kernel<<<(nz + BDIM - 1) / BDIM, BDIM, 0, stream>>>(wkv, nz);
    pass1_kv_kernel<<<BH * SPLIT, BDIM, 0, stream>>>(k, v, mask, wkv, wks);
    pass2_out_kernel<<<BH * (T_ / TM), BDIM, 0, stream>>>(q, wkv, wks, out);
}